// TopicVerbAttention_72361609003621
// MI455X (gfx1250) — compile-verified
//
#include <hip/hip_runtime.h>
#include <math.h>

// ---------------------------------------------------------------------------
// TopicVerbAttention for MI455X (gfx1250), memory-roofline design.
//   Fold:  kq = k_w^T (q_w @ topic_emb + q_b); k_b term cancels in softmax.
//   Span:  s[l] = value[l]·kq / 16 ; masked softmax ; out = attn @ value
// Hot kernel is a 157 MB gather -> async tensor path (global_load_async_to_
// lds_b128) with per-wave double buffering, waits via s_wait_asynccnt.
// Precompute kq uses V_WMMA_F32_16X16X4_F32 (19 tile-waves, K-loop of 64).
// ---------------------------------------------------------------------------

#define D_WORD   300
#define D_TOPIC  256
#define LMAX     8
#define ROW_BYTES 1200                      // 300 f32 = 75 x 16B chunks
#define SPAN_FLOATS (LMAX * D_WORD)         // 2400
#define SPAN_BYTES  (LMAX * ROW_BYTES)      // 9600
#define WAVES_PER_BLOCK 2
#define SPANS_PER_WAVE  4
#define KQ_TILES 19                         // ceil(300/16)

typedef __attribute__((ext_vector_type(2))) float v2f;
typedef __attribute__((ext_vector_type(8))) float v8f;

// ---------------- kernel 1: kq = k_w^T (q_w @ topic_emb + q_b) via WMMA ----
__global__ __launch_bounds__(KQ_TILES * 32) void precompute_kq_wmma(
    const float* __restrict__ topic_emb, const float* __restrict__ k_w,
    const float* __restrict__ q_w, const float* __restrict__ q_b,
    float* __restrict__ kq)
{
  __shared__ float q_sh[D_TOPIC];
  const int tid = threadIdx.x;

  // phase 1: query = q_w @ topic_emb + q_b  (256 threads, plain matvec)
  if (tid < D_TOPIC) {
    float acc = q_b[tid];
    for (int j = 0; j < D_TOPIC; ++j) acc += q_w[tid * D_TOPIC + j] * topic_emb[j];
    q_sh[tid] = acc;
  }
  __syncthreads();

  // phase 2: kq tile per wave via v_wmma_f32_16x16x4_f32
  const int wid  = tid >> 5;
  const int lane = tid & 31;
  const int half = lane >> 4;       // K half selector
  const int m    = lane & 15;       // A-matrix row within tile
  const int d0   = wid * 16;        // kq tile base
  const int d    = d0 + m;

  v8f c = {};
  for (int k0 = 0; k0 < D_TOPIC; k0 += 4) {
    const int ka = k0 + 2 * half;
    v2f a, b;
    // A[m, k] = k_w^T[d, k] = k_w[k*300 + d], zero-padded past d=299
    a.x = (d < D_WORD) ? k_w[ka * D_WORD + d]       : 0.f;
    a.y = (d < D_WORD) ? k_w[(ka + 1) * D_WORD + d] : 0.f;
    // B[k, n] = query[k] replicated across all 16 columns
    b.x = q_sh[ka];
    b.y = q_sh[ka + 1];
    c = __builtin_amdgcn_wmma_f32_16x16x4_f32(
        /*neg_a=*/false, a, /*neg_b=*/false, b,
        /*c_mod=*/(short)0, c, /*reuse_a=*/false, /*reuse_b=*/false);
  }
  // C/D layout: VGPR r -> M=r (lanes 0-15) / M=r+8 (lanes 16-31), N=lane%16.
  // Every column holds the same value; lane 0 / lane 16 write their 8 rows.
  if ((lane & 15) == 0) {
    const int base = d0 + 8 * half;
#pragma unroll
    for (int r = 0; r < 8; ++r) {
      const int dd = base + r;
      if (dd < D_WORD) kq[dd] = c[r];
    }
  }
}

// wave32 all-lanes XOR-butterfly sum via DS_SWIZZLE (no LDS memory touched)
__device__ __forceinline__ float wave_sum32(float x) {
  x += __int_as_float(__builtin_amdgcn_ds_swizzle(__float_as_int(x), 0x401f)); // xor 16
  x += __int_as_float(__builtin_amdgcn_ds_swizzle(__float_as_int(x), 0x201f)); // xor 8
  x += __int_as_float(__builtin_amdgcn_ds_swizzle(__float_as_int(x), 0x101f)); // xor 4
  x += __int_as_float(__builtin_amdgcn_ds_swizzle(__float_as_int(x), 0x081f)); // xor 2
  x += __int_as_float(__builtin_amdgcn_ds_swizzle(__float_as_int(x), 0x041f)); // xor 1
  return x;
}

// ---------------- kernel 2: gather + fused attention -----------------------
__global__ __launch_bounds__(WAVES_PER_BLOCK * 32) void span_attn(
    const int* __restrict__ verb_ids, const int* __restrict__ span_len,
    const float* __restrict__ V, const float* __restrict__ kq,
    float* __restrict__ out, int N)
{
  // [wave][double-buffer][8 rows x 300 f32]  = 38400 B static LDS
  __shared__ float smem[WAVES_PER_BLOCK * 2 * SPAN_FLOATS];

  const int lane = threadIdx.x & 31;
  const int wid  = threadIdx.x >> 5;
  float* wbuf = &smem[wid * 2 * SPAN_FLOATS];

  // wave-relative LDS byte offset (low 32 bits of generic LDS pointer)
  const unsigned lds_lane = (unsigned)(uintptr_t)wbuf + (unsigned)(lane * 16);
  const unsigned long long v_lane =
      (unsigned long long)(uintptr_t)V + (unsigned)(lane * 16);

  // kq resident in registers, zero-padded past column 299
  float kqr[10];
#pragma unroll
  for (int c = 0; c < 10; ++c) {
    const int d = lane + 32 * c;
    kqr[c] = (d < D_WORD) ? kq[d] : 0.f;
  }

  const int n0 = (blockIdx.x * WAVES_PER_BLOCK + wid) * SPANS_PER_WAVE;
  if (n0 >= N) return;

  // ---- stage one span's 8 value rows into LDS buffer `sel` (async path)
  auto issue = [&](int n, int sel) {
    const int* ids = verb_ids + n * LMAX;
    const unsigned lbuf = lds_lane + (unsigned)(sel * SPAN_BYTES);
#pragma unroll
    for (int l = 0; l < LMAX; ++l) {
      const unsigned long long ga =
          v_lane + (unsigned long long)((unsigned)ids[l] * (unsigned)ROW_BYTES);
      const unsigned la = lbuf + (unsigned)(l * ROW_BYTES);
      // chunks 0..31, 32..63, 64..74 (offset applies to BOTH lds+global addr)
      asm volatile("global_load_async_to_lds_b128 %0, %1, off"
                   :: "v"(la), "v"(ga) : "memory");
      asm volatile("global_load_async_to_lds_b128 %0, %1, off offset:512"
                   :: "v"(la), "v"(ga) : "memory");
      if (lane < 11)
        asm volatile("global_load_async_to_lds_b128 %0, %1, off offset:1024"
                     :: "v"(la), "v"(ga) : "memory");
    }
  };

  // ---- dot / softmax / weighted-sum from LDS buffer `sel`
  auto compute = [&](int n, int sel) {
    const float* buf = wbuf + sel * SPAN_FLOATS;
    float s[LMAX];
#pragma unroll
    for (int l = 0; l < LMAX; ++l) {
      float acc = 0.f;
#pragma unroll
      for (int c = 0; c < 10; ++c) {
        const int d = lane + 32 * c;
        const float v = (d < D_WORD) ? buf[l * D_WORD + d] : 0.f;
        acc += v * kqr[c];
      }
      s[l] = wave_sum32(acc) * 0.0625f;   // 1/sqrt(256)
    }
    const int slen = span_len[n];
    float mx = -INFINITY;
#pragma unroll
    for (int l = 0; l < LMAX; ++l) if (l < slen) mx = fmaxf(mx, s[l]);
    float e[LMAX];
    float sum = 0.f;
#pragma unroll
    for (int l = 0; l < LMAX; ++l) {
      e[l] = (l < slen) ? __expf(s[l] - mx) : 0.f;
      sum += e[l];
    }
    const float inv = 1.f / sum;
    float* o = out + (size_t)n * D_WORD;
#pragma unroll
    for (int c = 0; c < 10; ++c) {
      const int d = lane + 32 * c;
      if (d < D_WORD) {
        float acc = 0.f;
#pragma unroll
        for (int l = 0; l < LMAX; ++l) acc += e[l] * buf[l * D_WORD + d];
        o[d] = acc * inv;   // coalesced f32 store
      }
    }
  };

  // ---- software pipeline: overlap next span's gather with current compute
  issue(n0, 0);
#pragma unroll
  for (int i = 0; i < SPANS_PER_WAVE; ++i) {
    const int n = n0 + i;
    if (n >= N) break;
    if (i + 1 < SPANS_PER_WAVE && (n + 1) < N) {
      // LDS reads of the buffer we are about to overwrite must have retired
      asm volatile("s_wait_dscnt 0x0" ::: "memory");
      issue(n + 1, (i + 1) & 1);
      // 24 newer ops in flight; <=24 outstanding => span i fully landed
      asm volatile("s_wait_asynccnt 0x18" ::: "memory");
    } else {
      asm volatile("s_wait_asynccnt 0x0" ::: "memory");
    }
    compute(n, i & 1);
  }
}

// ---------------------------------------------------------------------------
extern "C" void kernel_launch(void* const* d_in, const int* in_sizes, int n_in,
                              void* d_out, int out_size, void* d_ws, size_t ws_size,
                              hipStream_t stream) {
  const float* topic_emb = (const float*)d_in[0];
  const int*   verb_ids  = (const int*)d_in[1];
  const int*   span_len  = (const int*)d_in[2];
  const float* V         = (const float*)d_in[3];
  const float* k_w       = (const float*)d_in[4];
  // d_in[5] = k_b : provably cancels inside softmax (constant shift), unused
  const float* q_w       = (const float*)d_in[6];
  const float* q_b       = (const float*)d_in[7];
  float* out = (float*)d_out;
  float* kq  = (float*)d_ws;          // 300 floats of scratch

  const int N = in_sizes[2];          // number of spans (span_len count)

  precompute_kq_wmma<<<1, KQ_TILES * 32, 0, stream>>>(topic_emb, k_w, q_w, q_b, kq);

  const int spans_per_block = WAVES_PER_BLOCK * SPANS_PER_WAVE;  // 8
  const int grid = (N + spans_per_block - 1) / spans_per_block;  // 2048 for N=16384
  span_attn<<<grid, WAVES_PER_BLOCK * 32, 0, stream>>>(
      verb_ids, span_len, V, kq, out, N);
}